// kernel_manual_1_21887153341179
// MI455X (gfx1250) — compile-verified
//
#include <hip/hip_runtime.h>

// ---------------------------------------------------------------------------
// Fully fused:  y(128,28,28) = conv3x3( pairwise-mixed depthwise(x), w1 )
//
//   mixed[c=2l+kk, h, w] = sum_{p=0,1} sum_{i=0..2} x[2l+p,h,w+i-1]*w2[2l+p,i,kk]
//   y[o,h,w]             = sum_{c,r,s} mixed[c,h+r-1,w+s-1] * w1[o,c>>1,r,s,c&1]
//
// GEMM view: M=128 (o), N=784 (h*28+w), K=576 (rs-major: k = (r*3+s)*64 + c).
// ~116 MFLOP, ~1 MB data -> entirely L2 resident -> latency bound. Therefore:
//   * ONE kernel launch (stage 1 recomputed per block into LDS, ~46K MAC/blk)
//   * zero-padding halo baked into LDS -> hot loop has NO masks / NO global
//     B loads: 1x ds_load_b64 (B frag) + 1x global b64 (A frag) per v_wmma
//   * fp32 WMMA (V_WMMA_F32_16X16X4_F32) keeps full reference precision
// Block = 256 thr = 8 wave32 = all 8 M-tiles of one N-tile (B reused 8x).
// Grid  = 49 blocks (784/16 N-tiles).
// ---------------------------------------------------------------------------

typedef float v2f __attribute__((ext_vector_type(2)));
typedef float v8f __attribute__((ext_vector_type(8)));

#define LDS_ELEMS (32 * 4 * 30 * 2)   // [cpair][row 0..3][win 0..29][kk] = 7680 f32

__global__ __launch_bounds__(256) void fused_conv_wmma(
    const float* __restrict__ x,    // (64,28,28)
    const float* __restrict__ w2,   // (64,3,2)
    const float* __restrict__ w1,   // (128,32,3,3,2)
    float* __restrict__ out)        // (128,28,28)
{
    __shared__ float btile[LDS_ELEMS];          // 30 KB

    const int tileN = blockIdx.x * 16;          // 0,16,...,768
    const int h0    = tileN / 28;               // min h touched by this N-tile

    // ------------- Phase A: stage-1 slice -> LDS (zero halo baked in) ------
    // LDS element t: kk=t&1, wi=(t>>1)%30, ri=(t/60)%4, cp=t/240
    //   holds mixed[c=2*cp+kk, hin=h0-1+ri, win=wi-1]  (0 outside the image)
    for (int t = threadIdx.x; t < LDS_ELEMS; t += 256) {
        const int kk = t & 1;
        const int wi = (t >> 1) % 30;
        const int ri = (t / 60) % 4;
        const int cp = t / 240;
        const int hin = h0 - 1 + ri;
        const int win = wi - 1;

        float v = 0.0f;
        if (hin >= 0 && hin < 28 && win >= 0 && win < 28) {
            const int j0 = cp * 2;
#pragma unroll
            for (int p = 0; p < 2; ++p) {
                const float* xrow = x + (j0 + p) * 784 + hin * 28;
                const float* w2r  = w2 + (j0 + p) * 6 + kk;
#pragma unroll
                for (int i = 0; i < 3; ++i) {
                    const int wx = win + i - 1;
                    const float xv = (wx >= 0 && wx < 28) ? xrow[wx] : 0.0f;
                    v += xv * w2r[i * 2];
                }
            }
        }
        btile[t] = v;
    }
    __syncthreads();

    // ------------- Phase B: implicit GEMM, one 16x16 tile per wave ---------
    const int lane    = threadIdx.x & 31;
    const int wave    = threadIdx.x >> 5;
    const int tileM   = wave * 16;              // 8 waves cover M = 0..127
    const int halfsel = lane >> 4;              // 0 -> K pair {0,1}, 1 -> {2,3}
    const int lane16  = lane & 15;

    const int mRow = tileM + lane16;            // A-fragment row (weights)
    const int nCol = tileN + lane16;            // B-fragment column (pixel)
    const int h = nCol / 28;
    const int w = nCol - h * 28;

    v8f acc = {};

#pragma unroll 3
    for (int rs = 0; rs < 9; ++rs) {
        const int r = rs / 3, s = rs - r * 3;
        // w1 flat: o*576 + l*18 + r*6 + s*2 + k ; element offset always even
        const int aBase = mRow * 576 + r * 6 + s * 2;
        // LDS: pairs (c, c+1) contiguous -> one aligned b64 per fragment reg
        //   off = ((cp*4 + (h+r-h0))*30 + (w+s)) * 2
        const int bRow = ((h + r - h0) * 30 + (w + s)) * 2;

        for (int c0 = 0; c0 < 64; c0 += 4) {
            const int cA = c0 + halfsel * 2;    // even channel for this lane
            const v2f a = *reinterpret_cast<const v2f*>(
                w1 + aBase + (cA >> 1) * 18);
            const v2f b = *reinterpret_cast<const v2f*>(
                btile + (cA >> 1) * 240 + bRow);
            acc = __builtin_amdgcn_wmma_f32_16x16x4_f32(
                /*neg_a=*/false, a, /*neg_b=*/false, b,
                /*c_mod=*/(short)0, acc, /*reuse_a=*/false, /*reuse_b=*/false);
        }
    }

    // C/D layout: VGPR rr -> M = tileM + 8*halfsel + rr, N = tileN + lane16
    const int mBase = tileM + halfsel * 8;
#pragma unroll
    for (int rr = 0; rr < 8; ++rr) {
        out[(mBase + rr) * 784 + tileN + lane16] = acc[rr];
    }
}

// ---------------------------------------------------------------------------
extern "C" void kernel_launch(void* const* d_in, const int* in_sizes, int n_in,
                              void* d_out, int out_size, void* d_ws, size_t ws_size,
                              hipStream_t stream) {
    const float* x  = (const float*)d_in[0];   // (1,64,28,28)
    const float* w1 = (const float*)d_in[1];   // (128,32,3,3,2)
    const float* w2 = (const float*)d_in[2];   // (64,3,2)
    float* out = (float*)d_out;                // (1,128,28,28)

    // Single fused launch: 49 N-tiles, 8 wave32 (8 M-tiles) per block.
    fused_conv_wmma<<<49, 256, 0, stream>>>(x, w2, w1, out);
}